// Attention_8203387536068
// MI455X (gfx1250) — compile-verified
//
#include <hip/hip_runtime.h>
#include <hip/hip_bf16.h>
#include <math.h>

typedef __attribute__((ext_vector_type(2))) float v2f;
typedef __attribute__((ext_vector_type(4))) float v4f;
typedef __attribute__((ext_vector_type(8))) float v8f;

#define B_    2
#define H_    16
#define S_    2048
#define D_    64
#define QROWS 16                 // query rows per wave
#define WAVES 8                  // waves per workgroup
#define QBLK  (QROWS * WAVES)    // 128 query rows per workgroup
#define KTILE 64                 // keys (or values) staged in LDS per iteration

// One wave owns a 16-row query tile. Two passes:
//  pass 1: raw scores -> attention buffer (L2-resident), online (m,l)
//  pass 2: normalize in place (NT stores) + P*V via WMMA
__global__ __launch_bounds__(256)
void attn_fused(const float* __restrict__ Q,
                const float* __restrict__ K,
                const float* __restrict__ V,
                const unsigned char* __restrict__ mask,   // jnp bool = 1 byte/elem, [B,S]
                float* __restrict__ att,                  // [B,H,S,S]
                float* __restrict__ outv)                 // [B,H,S,D]
{
    __shared__ float lds_tile[KTILE * D_];     // 16 KB: K tile in pass 1, V tile in pass 2
    __shared__ float lds_crow[WAVES * QROWS];  // per-row m + ln(l)

    const int tid  = threadIdx.x;
    const int wave = tid >> 5;
    const int lane = tid & 31;
    const int m    = lane & 15;   // A-row / B-C-D column index
    const int hi   = lane >> 4;

    const int bh   = blockIdx.x / (S_ / QBLK);
    const int qblk = blockIdx.x % (S_ / QBLK);
    const int bidx = bh / H_;
    const int q0   = qblk * QBLK + wave * QROWS;

    const float* Qb = Q + (size_t)bh * S_ * D_;
    const float* Kb = K + (size_t)bh * S_ * D_;
    const float* Vb = V + (size_t)bh * S_ * D_;
    float* attb = att + (size_t)bh * S_ * S_;

    // ---- Q tile in registers, A-matrix swizzle, pre-scaled by 1/sqrt(D)=0.125
    v2f qa[16];
    {
        const float* qrow = Qb + (size_t)(q0 + m) * D_;
        #pragma unroll
        for (int t = 0; t < 16; ++t) {
            v2f v = *(const v2f*)(qrow + 4 * t + 2 * hi);
            qa[t].x = v.x * 0.125f;
            qa[t].y = v.y * 0.125f;
        }
    }

    float mrun[8], lrun[8];
    #pragma unroll
    for (int r = 0; r < 8; ++r) { mrun[r] = -INFINITY; lrun[r] = 0.0f; }

    // ================= PASS 1: raw scores + online softmax stats =================
    for (int kt0 = 0; kt0 < S_; kt0 += KTILE) {
        { // cooperative coalesced stage of K tile: 64 keys x 64 dims = 16 KB
            const v4f* src = (const v4f*)(Kb + (size_t)kt0 * D_);
            v4f* dst = (v4f*)lds_tile;
            #pragma unroll
            for (int i = 0; i < 4; ++i)
                dst[i * 256 + tid] = src[i * 256 + tid];
        }
        __syncthreads();

        #pragma unroll
        for (int sub = 0; sub < 4; ++sub) {
            const int kbase = sub * 16;
            v8f c = {};
            #pragma unroll
            for (int t = 0; t < 16; ++t) {   // accumulate over D=64 in K=4 steps
                v2f b = *(const v2f*)(lds_tile + (kbase + m) * D_ + 4 * t + 2 * hi);
                c = __builtin_amdgcn_wmma_f32_16x16x4_f32(false, qa[t], false, b,
                                                          (short)0, c, false, false);
            }
            // reference semantics: mask True -> -1e30 at key column kt0+kbase+n (n==m)
            const int kcol = kt0 + kbase + m;
            if (mask[bidx * S_ + kcol]) {
                #pragma unroll
                for (int r = 0; r < 8; ++r) c[r] = -1e30f;
            }
            #pragma unroll
            for (int r = 0; r < 8; ++r) {
                const int qr = q0 + r + 8 * hi;
                attb[(size_t)qr * S_ + kcol] = c[r];          // raw score, stays in L2
                // row-max across the 16 lanes holding this row
                float v = c[r];
                v = fmaxf(v, __shfl_xor(v, 1, 32));
                v = fmaxf(v, __shfl_xor(v, 2, 32));
                v = fmaxf(v, __shfl_xor(v, 4, 32));
                v = fmaxf(v, __shfl_xor(v, 8, 32));
                const float mn = fmaxf(mrun[r], v);
                float e = __expf(c[r] - mn);
                e += __shfl_xor(e, 1, 32);
                e += __shfl_xor(e, 2, 32);
                e += __shfl_xor(e, 4, 32);
                e += __shfl_xor(e, 8, 32);
                lrun[r] = lrun[r] * __expf(mrun[r] - mn) + e;
                mrun[r] = mn;
            }
        }
        __syncthreads();
    }

    // per-row coefficient c = m + ln(l); move from D-layout to A-layout via LDS
    if (m == 0) {
        #pragma unroll
        for (int r = 0; r < 8; ++r)
            lds_crow[wave * QROWS + r + 8 * hi] = mrun[r] + __logf(lrun[r]);
    }
    __syncthreads();
    const float crow = lds_crow[wave * QROWS + m];

    // ================= PASS 2: normalize in place + P*V =================
    v8f acc0 = {}, acc1 = {}, acc2 = {}, acc3 = {};
    float* srow = attb + (size_t)(q0 + m) * S_;

    for (int kt0 = 0; kt0 < S_; kt0 += KTILE) {
        { // stage V tile
            const v4f* src = (const v4f*)(Vb + (size_t)kt0 * D_);
            v4f* dst = (v4f*)lds_tile;
            #pragma unroll
            for (int i = 0; i < 4; ++i)
                dst[i * 256 + tid] = src[i * 256 + tid];
        }
        __syncthreads();

        #pragma unroll
        for (int t = 0; t < 16; ++t) {       // 4 keys per step
            const int kc = 4 * t + 2 * hi;   // key offset within tile for this lane
            v2f s = *(const v2f*)(srow + kt0 + kc);   // L2 hit (written in pass 1)
            v2f p;
            p.x = __expf(s.x - crow);
            p.y = __expf(s.y - crow);
            // final normalized attention: streamed, never re-read -> non-temporal
            __builtin_nontemporal_store(p.x, srow + kt0 + kc);
            __builtin_nontemporal_store(p.y, srow + kt0 + kc + 1);

            const float* v0 = lds_tile + kc * D_;        // V row kc
            const float* v1 = lds_tile + (kc + 1) * D_;  // V row kc+1
            v2f b0, b1, b2, b3;
            b0.x = v0[m];      b0.y = v1[m];
            b1.x = v0[16 + m]; b1.y = v1[16 + m];
            b2.x = v0[32 + m]; b2.y = v1[32 + m];
            b3.x = v0[48 + m]; b3.y = v1[48 + m];
            acc0 = __builtin_amdgcn_wmma_f32_16x16x4_f32(false, p, false, b0, (short)0, acc0, false, false);
            acc1 = __builtin_amdgcn_wmma_f32_16x16x4_f32(false, p, false, b1, (short)0, acc1, false, false);
            acc2 = __builtin_amdgcn_wmma_f32_16x16x4_f32(false, p, false, b2, (short)0, acc2, false, false);
            acc3 = __builtin_amdgcn_wmma_f32_16x16x4_f32(false, p, false, b3, (short)0, acc3, false, false);
        }
        __syncthreads();
    }

    float* ob = outv + (size_t)bh * S_ * D_;
    #pragma unroll
    for (int r = 0; r < 8; ++r) {
        const int qr = q0 + r + 8 * hi;
        __builtin_nontemporal_store(acc0[r], ob + (size_t)qr * D_ + m);
        __builtin_nontemporal_store(acc1[r], ob + (size_t)qr * D_ + 16 + m);
        __builtin_nontemporal_store(acc2[r], ob + (size_t)qr * D_ + 32 + m);
        __builtin_nontemporal_store(acc3[r], ob + (size_t)qr * D_ + 48 + m);
    }
}

extern "C" void kernel_launch(void* const* d_in, const int* in_sizes, int n_in,
                              void* d_out, int out_size, void* d_ws, size_t ws_size,
                              hipStream_t stream) {
    const float* q = (const float*)d_in[0];
    const float* k = (const float*)d_in[1];
    const float* v = (const float*)d_in[2];
    const unsigned char* mask = (const unsigned char*)d_in[3];  // jnp bool = 1 byte

    float* att  = (float*)d_out;                                  // [B,H,S,S]
    float* outv = att + (size_t)B_ * H_ * S_ * S_;                // [B,H,S,D]

    const int grid = B_ * H_ * (S_ / QBLK);   // 2*16*16 = 512 workgroups
    attn_fused<<<grid, 256, 0, stream>>>(q, k, v, mask, att, outv);
}